// PRVBranch_53678501265970
// MI455X (gfx1250) — compile-verified
//
#include <hip/hip_runtime.h>
#include <hip/hip_bf16.h>
#include <math.h>

// ---------------- problem constants ----------------
#define ROWLEN 9000
#define NW 599          // (9000-30)/15 + 1 windows
#define NR 598          // rr intervals
#define WSZ 30
#define STRD 15

typedef __attribute__((ext_vector_type(2))) float v2f;
typedef __attribute__((ext_vector_type(8))) float v8f;

// ============================================================
// Kernel 1: per-row feature extraction (memory-bound pass)
// One block (256 thr / 8 waves) per batch row; row staged in LDS.
// Emits feats[row][5] = {mean_rr, rmssd, sdnn, lf_power, hf_power}
// ============================================================
__global__ __launch_bounds__(256) void prv_feats_kernel(const float* __restrict__ x,
                                                        float* __restrict__ feats) {
    __shared__ float s_x[ROWLEN];
    __shared__ float s_pk[NW];
    __shared__ float s_rr[NR];
    __shared__ float s_red[8][28];

    const int t = threadIdx.x;
    const int row = blockIdx.x;

    // ---- stage the row into LDS with 128-bit loads ----
    const float4* xr = (const float4*)(x + (size_t)row * ROWLEN);
    float4* sx4 = (float4*)s_x;
    for (int i = t; i < ROWLEN / 4; i += 256) sx4[i] = xr[i];
    __syncthreads();

    // ---- soft-argmax per window -> peak positions ----
    for (int w = t; w < NW; w += 256) {
        const float* win = s_x + w * STRD;
        float m = win[0];
        #pragma unroll
        for (int i = 1; i < WSZ; ++i) m = fmaxf(m, win[i]);
        float z = 0.f, e = 0.f;
        #pragma unroll
        for (int i = 0; i < WSZ; ++i) {
            float p = __expf((win[i] - m) * 10.0f);   // 1/TEMP = 10
            z += p;
            e += p * (float)i;
        }
        s_pk[w] = e / z + (float)(w * STRD);
    }
    __syncthreads();

    // ---- RR intervals ----
    for (int i = t; i < NR; i += 256) s_rr[i] = (s_pk[i + 1] - s_pk[i]) * (1.0f / 30.0f);
    __syncthreads();

    // ---- partial sums: S1, S2, sum(diff^2), 12 DFT bins (k=2..13 of a 1024-pt rfft) ----
    float S1 = 0.f, S2 = 0.f, SD = 0.f;
    float re[12], im[12];
    #pragma unroll
    for (int k = 0; k < 12; ++k) { re[k] = 0.f; im[k] = 0.f; }
    const float omega = 2.0f * 3.14159265358979323846f / 1024.0f;
    for (int j = t; j < NR; j += 256) {
        float v = s_rr[j];
        S1 += v; S2 += v * v;
        if (j < NR - 1) { float d = s_rr[j + 1] - v; SD += d * d; }
        float s1, c1;
        __sincosf(omega * (float)j, &s1, &c1);
        float c = c1 * c1 - s1 * s1;     // angle 2*theta (k starts at 2)
        float s = 2.f * s1 * c1;
        #pragma unroll
        for (int k = 0; k < 12; ++k) {
            re[k] += v * c; im[k] += v * s;
            float cn = c * c1 - s * s1;  // rotate by theta -> next k
            float sn = s * c1 + c * s1;
            c = cn; s = sn;
        }
    }

    // ---- block reduction of 27 scalars: wave shfl_xor, then cross-wave via LDS ----
    float vals[27];
    vals[0] = S1; vals[1] = S2; vals[2] = SD;
    #pragma unroll
    for (int k = 0; k < 12; ++k) { vals[3 + k] = re[k]; vals[15 + k] = im[k]; }
    #pragma unroll
    for (int q = 0; q < 27; ++q) {
        float v = vals[q];
        #pragma unroll
        for (int off = 16; off >= 1; off >>= 1) v += __shfl_xor(v, off, 32);
        vals[q] = v;
    }
    const int wave = t >> 5, lane = t & 31;
    if (lane == 0)
        for (int q = 0; q < 27; ++q) s_red[wave][q] = vals[q];
    __syncthreads();

    if (t == 0) {
        float tot[27];
        for (int q = 0; q < 27; ++q) {
            float a = 0.f;
            for (int wv = 0; wv < 8; ++wv) a += s_red[wv][q];
            tot[q] = a;
        }
        const float n = (float)NR;
        float mean  = tot[0] / n;
        float var   = (tot[1] - n * mean * mean) / (n - 1.0f);   // ddof=1
        float sdnn  = sqrtf(fmaxf(var, 0.f));
        float rmssd = sqrtf(tot[2] / (n - 1.0f) + 1e-6f);        // 597 diffs
        float lf = 0.f, hf = 0.f;
        for (int k = 0; k < 4;  ++k) lf += tot[3 + k] * tot[3 + k] + tot[15 + k] * tot[15 + k];
        for (int k = 4; k < 12; ++k) hf += tot[3 + k] * tot[3 + k] + tot[15 + k] * tot[15 + k];
        float* f = feats + (size_t)row * 5;
        f[0] = mean; f[1] = rmssd; f[2] = sdnn; f[3] = lf; f[4] = hf;
    }
}

// ============================================================
// Kernel 2: MLP + LayerNorm with V_WMMA_F32_16X16X4_F32.
// One wave per 16-row tile; 4 waves per block. FP32 throughout
// (matches reference precision; K=5 / K=64 fit the x4 shape).
// ============================================================
__global__ __launch_bounds__(128) void prv_mlp_ln_kernel(
    const float* __restrict__ feats,
    const float* __restrict__ W1, const float* __restrict__ b1,
    const float* __restrict__ W2, const float* __restrict__ b2,
    const float* __restrict__ gamma, const float* __restrict__ beta,
    float* __restrict__ out) {

    __shared__ float s_h[4][16 * 64];     // per-wave h staging (C-layout -> A-layout)

    const int lane = threadIdx.x & 31;
    const int wave = threadIdx.x >> 5;
    const int tile = blockIdx.x * 4 + wave;
    const int half = lane >> 4;           // 0: lanes 0-15, 1: lanes 16-31
    const int l16  = lane & 15;
    const int row0 = tile * 16;

    // ---- layer 1: (16x5) @ (5x64), K padded to 8 -> 2 wmma k-steps ----
    // A 16x4 f32 layout: lanes 0-15 hold K={0,1}, lanes 16-31 hold K={2,3}
    const float* fr = feats + (size_t)(row0 + l16) * 5;
    v2f a0, a1;
    a0.x = fr[2 * half + 0];
    a0.y = fr[2 * half + 1];
    a1.x = half ? 0.f : fr[4];            // K=4 is the only valid lane of chunk 1
    a1.y = 0.f;

    float* sh = s_h[wave];
    #pragma unroll
    for (int nt = 0; nt < 4; ++nt) {
        const int col = nt * 16 + l16;
        v2f b0, bx;
        b0.x = W1[(2 * half + 0) * 64 + col];
        b0.y = W1[(2 * half + 1) * 64 + col];
        bx.x = half ? 0.f : W1[4 * 64 + col];
        bx.y = 0.f;
        v8f c = {};
        c = __builtin_amdgcn_wmma_f32_16x16x4_f32(false, a0, false, b0, (short)0, c, false, false);
        c = __builtin_amdgcn_wmma_f32_16x16x4_f32(false, a1, false, bx, (short)0, c, false, false);
        const float bias = b1[col];
        #pragma unroll
        for (int r = 0; r < 8; ++r) {
            // D layout: lane half 0 holds rows 0-7, half 1 holds rows 8-15
            sh[(r + 8 * half) * 64 + col] = fmaxf(c[r] + bias, 0.f);
        }
    }
    __syncthreads();

    // ---- layer 2: (16x64) @ (64x128) -> 16 k-steps x 8 n-tiles ----
    v8f acc[8];
    #pragma unroll
    for (int nt = 0; nt < 8; ++nt) acc[nt] = (v8f){};
    for (int kc = 0; kc < 16; ++kc) {
        const int k = kc * 4 + 2 * half;
        v2f a;
        a.x = sh[l16 * 64 + k + 0];
        a.y = sh[l16 * 64 + k + 1];
        #pragma unroll
        for (int nt = 0; nt < 8; ++nt) {
            const int col = nt * 16 + l16;
            v2f b;
            b.x = W2[(k + 0) * 128 + col];
            b.y = W2[(k + 1) * 128 + col];
            acc[nt] = __builtin_amdgcn_wmma_f32_16x16x4_f32(false, a, false, b, (short)0,
                                                            acc[nt], false, false);
        }
    }

    // ---- bias + LayerNorm directly from D-matrix layout ----
    float s[8], q[8];
    #pragma unroll
    for (int r = 0; r < 8; ++r) { s[r] = 0.f; q[r] = 0.f; }
    #pragma unroll
    for (int nt = 0; nt < 8; ++nt) {
        const float bias = b2[nt * 16 + l16];
        #pragma unroll
        for (int r = 0; r < 8; ++r) {
            float v = acc[nt][r] + bias;
            acc[nt][r] = v;
            s[r] += v; q[r] += v * v;
        }
    }
    // each row's 128 cols live across 16 lanes within a half-wave
    #pragma unroll
    for (int r = 0; r < 8; ++r) {
        float sv = s[r], qv = q[r];
        #pragma unroll
        for (int off = 1; off < 16; off <<= 1) {
            sv += __shfl_xor(sv, off, 32);
            qv += __shfl_xor(qv, off, 32);
        }
        s[r] = sv; q[r] = qv;
    }
    #pragma unroll
    for (int nt = 0; nt < 8; ++nt) {
        const int col = nt * 16 + l16;
        const float g = gamma[col], bt = beta[col];
        #pragma unroll
        for (int r = 0; r < 8; ++r) {
            float mean = s[r] * (1.0f / 128.0f);
            float var  = q[r] * (1.0f / 128.0f) - mean * mean;
            float inv  = rsqrtf(var + 1e-5f);
            out[(size_t)(row0 + r + 8 * half) * 128 + col] = (acc[nt][r] - mean) * inv * g + bt;
        }
    }
}

extern "C" void kernel_launch(void* const* d_in, const int* in_sizes, int n_in,
                              void* d_out, int out_size, void* d_ws, size_t ws_size,
                              hipStream_t stream) {
    const float* x     = (const float*)d_in[0];
    const float* W1    = (const float*)d_in[1];
    const float* b1    = (const float*)d_in[2];
    const float* W2    = (const float*)d_in[3];
    const float* b2    = (const float*)d_in[4];
    const float* gamma = (const float*)d_in[5];
    const float* beta  = (const float*)d_in[6];

    float* feats = (float*)d_ws;          // 4096 x 5 f32 scratch

    prv_feats_kernel<<<4096, 256, 0, stream>>>(x, feats);
    prv_mlp_ln_kernel<<<64, 128, 0, stream>>>(feats, W1, b1, W2, b2, gamma, beta,
                                              (float*)d_out);
}